// LatticeConvolution_3582002724892
// MI455X (gfx1250) — compile-verified
//
#include <hip/hip_runtime.h>
#include <hip/hip_bf16.h>

typedef __attribute__((ext_vector_type(2))) float v2f;
typedef __attribute__((ext_vector_type(8))) float v8f;

#define BATCH   256
#define CIN     8
#define NPTS    4096
#define KNBR    9
#define NSITE   2
#define COUT    8
#define RED     (CIN * KNBR)          // 72 reduction length
#define NCHUNK  (RED / 4)             // 18 WMMA K-chunks of 4
#define ROWLEN  (BATCH * CIN)         // 2048 floats per xT row

// ---------------------------------------------------------------------------
// Pass 1: transpose x (2048 x 4096) -> xT (4096 x 2048) so that the per-point
// neighbor gather reads contiguous row slices instead of stride-16KB dwords.
// ---------------------------------------------------------------------------
__global__ __launch_bounds__(256)
void lattice_transpose_kernel(const float* __restrict__ x, float* __restrict__ xT) {
    __shared__ float tile[32][33];
    const int tx = threadIdx.x;       // 0..31
    const int ty = threadIdx.y;       // 0..7
    const int n0 = blockIdx.x * 32;   // column tile in x (n dimension)
    const int r0 = blockIdx.y * 32;   // row tile in x (b*CIN + c dimension)

    #pragma unroll
    for (int i = 0; i < 32; i += 8) {
        tile[ty + i][tx] = x[(size_t)(r0 + ty + i) * NPTS + (n0 + tx)];
    }
    __syncthreads();
    #pragma unroll
    for (int i = 0; i < 32; i += 8) {
        xT[(size_t)(n0 + ty + i) * ROWLEN + (r0 + tx)] = tile[tx][ty + i];
    }
}

// ---------------------------------------------------------------------------
// Pass 2: per point n, out[b,0:8,n] = A(16b x 72) * W_s(72 x 8) + bias_s via
// 18 V_WMMA_F32_16X16X4_F32 split over two accumulator chains for ILP.
// Both sites' B-fragments live in VGPRs (loaded once, in native B layout).
// B columns 8..15 hold duplicated (finite) weights instead of zeros: column d
// of B only affects column d of D, and D columns 8..15 are never stored, so
// no masking/predication is needed on the preload.
// Site selection is a uniform scalar branch so EXEC is all-1s at each WMMA.
// Block = 256 threads = 8 waves; wave w -> batches [b0, b0+16); block loops
// over 16 consecutive points.
// ---------------------------------------------------------------------------
__global__ __launch_bounds__(256)
void lattice_conv_wmma_kernel(const float* __restrict__ xT,
                              const int*   __restrict__ nbr,
                              const int*   __restrict__ sub_ids,
                              const float* __restrict__ weight,
                              const float* __restrict__ bias,
                              float*       __restrict__ out) {
    const int lane = threadIdx.x & 31;
    const int wave = threadIdx.x >> 5;
    const int d    = lane & 15;                 // D/B column = output channel
    const int dc   = d & 7;                     // clamped (in-bounds) channel
    const int hi   = lane >> 4;                 // lane half

    // ---- Preload both weight matrices into VGPRs in WMMA B-fragment layout:
    //      lane L holds column N=d=(L&15); vgpr j holds K-row t*4 + (L<16?j:2+j)
    //      Unconditional loads (columns 8..15 duplicate 0..7; see note above).
    v2f wr[NSITE][NCHUNK];
    #pragma unroll
    for (int s = 0; s < NSITE; ++s) {
        #pragma unroll
        for (int t = 0; t < NCHUNK; ++t) {
            #pragma unroll
            for (int j = 0; j < 2; ++j) {
                const int r = t * 4 + hi * 2 + j;   // K-row this vgpr holds
                const int k = r >> 3;               // neighbor slot
                const int c = r & 7;                // input channel
                wr[s][t][j] = weight[(((s * CIN + c) * KNBR) + k) * COUT + dc];
            }
        }
    }
    const float bv0 = bias[dc];
    const float bv1 = bias[COUT + dc];

    // batch sub-tile for this wave: 2 block-halves x 8 waves x 16 batches
    const int b0 = (blockIdx.x & 1) * 128 + wave * 16;
    const int pg = blockIdx.x >> 1;             // point group (16 points)

    // A-fragment base offset within an xT row (two consecutive floats / lane)
    const int a_off = ((b0 + (lane & 15)) << 3) + (hi << 1);

    for (int pi = 0; pi < 16; ++pi) {
        const int n = pg * 16 + pi;
        const int s = __builtin_amdgcn_readfirstlane(sub_ids[n]);   // uniform
        const int* nn = nbr + n * KNBR;

        int idx[KNBR];
        #pragma unroll
        for (int k = 0; k < KNBR; ++k) idx[k] = nn[k];

        // Stage all 18 A fragments first -> one clause of global_load_b64,
        // overlapped with the WMMA chains below.
        v2f af[NCHUNK];
        #pragma unroll
        for (int t = 0; t < NCHUNK; ++t) {
            const int k  = t >> 1;              // which neighbor row
            const int c0 = (t & 1) * 4;         // channel base of this K-chunk
            af[t] = *(const v2f*)(xT + (size_t)idx[k] * ROWLEN + a_off + c0);
        }

        v8f acc0 = {};
        v8f acc1 = {};
        if (s == 0) {
            #pragma unroll
            for (int t = 0; t < NCHUNK; t += 2) {
                acc0 = __builtin_amdgcn_wmma_f32_16x16x4_f32(
                           false, af[t],     false, wr[0][t],     (short)0, acc0, false, false);
                acc1 = __builtin_amdgcn_wmma_f32_16x16x4_f32(
                           false, af[t + 1], false, wr[0][t + 1], (short)0, acc1, false, false);
            }
        } else {
            #pragma unroll
            for (int t = 0; t < NCHUNK; t += 2) {
                acc0 = __builtin_amdgcn_wmma_f32_16x16x4_f32(
                           false, af[t],     false, wr[1][t],     (short)0, acc0, false, false);
                acc1 = __builtin_amdgcn_wmma_f32_16x16x4_f32(
                           false, af[t + 1], false, wr[1][t + 1], (short)0, acc1, false, false);
            }
        }

        const float bv = (s == 0) ? bv0 : bv1;
        if (d < COUT) {
            const int bbase = b0 + (hi << 3);   // M = i + 8*hi
            #pragma unroll
            for (int i = 0; i < 8; ++i) {
                out[((size_t)(bbase + i) * COUT + d) * NPTS + n] =
                    acc0[i] + acc1[i] + bv;
            }
        }
    }
}

extern "C" void kernel_launch(void* const* d_in, const int* in_sizes, int n_in,
                              void* d_out, int out_size, void* d_ws, size_t ws_size,
                              hipStream_t stream) {
    const float* x      = (const float*)d_in[0];
    const int*   nbr    = (const int*)d_in[1];
    const int*   sids   = (const int*)d_in[2];
    const float* weight = (const float*)d_in[3];
    const float* bias   = (const float*)d_in[4];
    float*       out    = (float*)d_out;
    float*       xT     = (float*)d_ws;        // 4096 * 2048 * 4 = 33.5 MB

    // Pass 1: transpose x into workspace
    lattice_transpose_kernel<<<dim3(NPTS / 32, ROWLEN / 32), dim3(32, 8), 0, stream>>>(x, xT);

    // Pass 2: WMMA lattice convolution
    // blocks = (B/128) * (N/16) = 2 * 256 = 512
    lattice_conv_wmma_kernel<<<(BATCH / 128) * (NPTS / 16), 256, 0, stream>>>(
        xT, nbr, sids, weight, bias, out);
}